// EUNN_85375359910276
// MI455X (gfx1250) — compile-verified
//
#include <hip/hip_runtime.h>
#include <math.h>

// ---------------------------------------------------------------------------
// EUNN (B=8192, H=2048, C=32) recast as a banded GEMM on the CDNA5 matrix pipe.
//
//   out = W * x   where W is the composition of 16 brick-wall rotation steps.
//   W is banded: out[h] depends on x[h-32 .. h+32] (complex).
//
//   K1: rotation coefficient tables from angles             (1 MB scratch)
//   K2: columns of W via basis propagation -> 256 band mats (2.25 MB scratch,
//       stored directly in WMMA B-fragment order)
//   K3: OUT(8192x4096) = X(8192x4096) * band(W)^T via v_wmma_f32_16x16x4_f32
// ---------------------------------------------------------------------------

typedef float v2f __attribute__((ext_vector_type(2)));
typedef float v8f __attribute__((ext_vector_type(8)));

#define Hh      2048
#define HRr     4096           // reals per row
#define NSTEP   16
#define NTILE   256            // output tiles of 8 complex (16 reals)
#define KBAND   144            // band reals per tile (72 complex)
#define MROW    (KBAND * 16)   // floats per tile matrix (2304)
#define XSTRIDE 390            // LDS row stride (390 % 64 == 6 -> conflict-free)
#define KSPAN   384            // 16 tiles * 16 reals + 128 halo reals

// ---------------- K1: rotation coefficients --------------------------------
__global__ void __launch_bounds__(256) eunn_coef(const float* __restrict__ ang,
                                                 float2* __restrict__ cAm, float2* __restrict__ cAn,
                                                 float2* __restrict__ cBm, float2* __restrict__ cBn) {
    int t = blockIdx.x * 256 + threadIdx.x;
    if (t >= NSTEP * 1024) return;
    int s = t >> 10, k = t & 1023;
    float p0 = ang[(2 * k) * 32 + 2 * s];
    float t0 = ang[(2 * k + 1) * 32 + 2 * s];
    float p1 = ang[(2 * k) * 32 + 2 * s + 1];
    float t1 = ang[(2 * k + 1) * 32 + 2 * s + 1];
    float cp0, sp0, ct0, st0, cp1, sp1, ct1, st1;
    sincosf(p0, &sp0, &cp0);
    sincosf(t0, &st0, &ct0);
    sincosf(p1, &sp1, &cp1);
    sincosf(t1, &st1, &ct1);
    int i0 = s * Hh + 2 * k;
    cAm[i0]     = make_float2(cp0 * ct0,  sp0 * ct0);
    cAm[i0 + 1] = make_float2(ct0,        0.f);
    cAn[i0]     = make_float2(-cp0 * st0, -sp0 * st0);
    cAn[i0 + 1] = make_float2(st0,        0.f);
    cBm[i0]     = make_float2(cp1 * ct1,  sp1 * ct1);
    cBm[i0 + 1] = make_float2(ct1,        0.f);
    cBn[i0]     = make_float2(-cp1 * st1, -sp1 * st1);
    cBn[i0 + 1] = make_float2(st1,        0.f);
}

// complex: up = u*mu + v*nu
__device__ inline float2 cmadd2(float2 u, float2 mu, float2 v, float2 nu) {
    float2 r;
    r.x = fmaf(-v.y, nu.y, fmaf(v.x, nu.x, fmaf(-u.y, mu.y, u.x * mu.x)));
    r.y = fmaf( v.y, nu.x, fmaf(v.x, nu.y, fmaf( u.y, mu.x, u.x * mu.y)));
    return r;
}

// ---------------- K2: W columns via basis propagation + scatter ------------
// WG w handles columns g in [16w, 16w+16). Window: absolute positions
// p = 16w-32+l, l in [0,80). Support of column g stays within [g-32,g+32].
// Band matrices are written in WMMA B-fragment order:
//   element (k, jr) of the 144x16 tile -> (k>>2)*64 + ((k>>1)&1)*32 + jr*2 + (k&1)
// so each complex tap is one aligned float4 and the GEMM reads one float2/lane.
__global__ void __launch_bounds__(256) eunn_cols(const float2* __restrict__ cAm, const float2* __restrict__ cAn,
                                                 const float2* __restrict__ cBm, const float2* __restrict__ cBn,
                                                 float* __restrict__ Mmat) {
    __shared__ float2 S[16 * 80];
    const int w = blockIdx.x;            // 0..127
    const int tid = threadIdx.x;
    const int pbase = 16 * w - 32;

    for (int i = tid; i < 16 * 80; i += 256) {
        int b = i / 80, l = i % 80;
        float2 v = make_float2(0.f, 0.f);
        if (l == b + 32) v.x = 1.f;      // e_g at p = g
        S[i] = v;
    }
    __syncthreads();

    for (int s = 0; s < NSTEP; ++s) {
        const float2* mA = cAm + s * Hh;
        const float2* nA = cAn + s * Hh;
        // sub-step A: pairs (even l, l+1); 40 pairs x 16 bases
        for (int q = tid; q < 640; q += 256) {
            int b = q & 15, pr = q >> 4;
            int l = 2 * pr;
            int h = (pbase + l + 2048) & 2047;           // even
            float2 u = S[b * 80 + l], v = S[b * 80 + l + 1];
            float2 up = cmadd2(u, mA[h],     v, nA[h]);
            float2 vp = cmadd2(v, mA[h + 1], u, nA[h + 1]);
            S[b * 80 + l]     = up;
            S[b * 80 + l + 1] = vp;
        }
        __syncthreads();
        const float2* mB = cBm + s * Hh;
        const float2* nB = cBn + s * Hh;
        // sub-step B: pairs (odd l, l+1), l in {-1,1,...,79}; zero halo
        for (int q = tid; q < 656; q += 256) {
            int b = q & 15, pr = q >> 4;                 // pr 0..40
            int l = 2 * pr - 1;                          // -1 .. 79
            int h = (pbase + l + 2048) & 2047;           // odd (>=1)
            float2 u = (l >= 0) ? S[b * 80 + l]     : make_float2(0.f, 0.f);
            float2 v = (l < 79) ? S[b * 80 + l + 1] : make_float2(0.f, 0.f);
            float2 up = cmadd2(u, mB[h - 1], v, nB[h - 1]);
            float2 vp = cmadd2(v, mB[h],     u, nB[h]);
            if (l >= 0) S[b * 80 + l]     = up;
            if (l < 79) S[b * 80 + l + 1] = vp;
        }
        __syncthreads();
    }

    // scatter W[h][g] into per-tile band matrices (B-fragment order)
    for (int i = tid; i < 16 * 80; i += 256) {
        int b = i / 80, l = i % 80;
        int d = l - 32 - b;                              // p - g
        if (d < -32 || d > 32) continue;
        int g = 16 * w + b;
        int p = pbase + l;
        int h = (p + 2048) & 2047;
        int n = h >> 3, j = h & 7;
        int delta = (g - 8 * n + 32 + 2048) & 2047;      // in [0,72)
        float2 wv = S[b * 80 + l];
        // rows k0=2*delta, k0+1; columns jr=2j, 2j+1  ->  one float4
        float* base = Mmat + n * MROW
                    + (delta >> 1) * 64 + (delta & 1) * 32 + 4 * j;
        *(float4*)base = make_float4(wv.x, -wv.y, wv.y, wv.x);
    }
}

// ---------------- K3: banded GEMM with V_WMMA_F32_16X16X4_F32 --------------
// grid (16, 256): x = group of 16 n-tiles, y = group of 32 batch rows.
__global__ void __launch_bounds__(256) eunn_gemm(const float* __restrict__ X,
                                                 const float* __restrict__ Mmat,
                                                 float* __restrict__ Out) {
    __shared__ float Xs[32 * XSTRIDE];
    const int ng = blockIdx.x;              // 0..15
    const int bg = blockIdx.y;              // 0..255
    const int n0 = ng * 16;
    const int r0 = 16 * n0 - 64;            // first band real (can be negative)
    const int rowbase = bg * 32;
    const int tid = threadIdx.x;

    // stage 32 rows x 384 band reals (mod-4096 ring gather), coalesced
    for (int idx = tid; idx < 32 * KSPAN; idx += 256) {
        int row = idx / KSPAN, c = idx % KSPAN;
        int r = (r0 + c + 4096) & 4095;
        Xs[row * XSTRIDE + c] = X[(rowbase + row) * HRr + r];
    }
    __syncthreads();

    const int wave = tid >> 5;              // wave32
    const int lane = tid & 31;
    const int Msel = lane & 15;
    const int hi = lane >> 4;               // K half select (0 -> K0/1, 1 -> K2/3)

    for (int task = wave; task < 32; task += 8) {
        int nt = task >> 1, ms = task & 1;
        int n = n0 + nt;
        const float2* Mt2 = (const float2*)(Mmat + n * MROW);  // fragment order
        const float* xrow = &Xs[(ms * 16 + Msel) * XSTRIDE + nt * 16];

        v8f acc0 = {0.f, 0.f, 0.f, 0.f, 0.f, 0.f, 0.f, 0.f};
        v8f acc1 = {0.f, 0.f, 0.f, 0.f, 0.f, 0.f, 0.f, 0.f};
        #pragma unroll 6
        for (int kb = 0; kb < 36; kb += 2) {
            {
                int k0 = kb * 4 + hi * 2;
                v2f a; a.x = xrow[k0]; a.y = xrow[k0 + 1];
                float2 bf = Mt2[kb * 32 + hi * 16 + Msel];   // one b64, coalesced
                v2f b; b.x = bf.x; b.y = bf.y;
                acc0 = __builtin_amdgcn_wmma_f32_16x16x4_f32(false, a, false, b,
                                                             (short)0, acc0, false, false);
            }
            {
                int k0 = (kb + 1) * 4 + hi * 2;
                v2f a; a.x = xrow[k0]; a.y = xrow[k0 + 1];
                float2 bf = Mt2[(kb + 1) * 32 + hi * 16 + Msel];
                v2f b; b.x = bf.x; b.y = bf.y;
                acc1 = __builtin_amdgcn_wmma_f32_16x16x4_f32(false, a, false, b,
                                                             (short)0, acc1, false, false);
            }
        }

        // D layout: VGPR r -> (M = r + hi*8, N = Msel); stream out non-temporal
        float* orow = Out + (rowbase + ms * 16 + hi * 8) * HRr + n * 16 + Msel;
        #pragma unroll
        for (int r = 0; r < 8; ++r)
            __builtin_nontemporal_store(acc0[r] + acc1[r], orow + r * HRr);
    }
}

// ---------------------------------------------------------------------------
extern "C" void kernel_launch(void* const* d_in, const int* in_sizes, int n_in,
                              void* d_out, int out_size, void* d_ws, size_t ws_size,
                              hipStream_t stream) {
    const float* x   = (const float*)d_in[0];   // (8192, 2048, 2) f32
    const float* ang = (const float*)d_in[1];   // (2048, 32) f32
    float* out = (float*)d_out;

    // workspace layout: 4 coefficient tables (float2[16*2048]) + band matrices
    float2* cAm = (float2*)d_ws;
    float2* cAn = cAm + NSTEP * Hh;
    float2* cBm = cAn + NSTEP * Hh;
    float2* cBn = cBm + NSTEP * Hh;
    float*  Mmat = (float*)(cBn + NSTEP * Hh);

    hipMemsetAsync(Mmat, 0, (size_t)NTILE * MROW * sizeof(float), stream);
    eunn_coef<<<64, 256, 0, stream>>>(ang, cAm, cAn, cBm, cBn);
    eunn_cols<<<128, 256, 0, stream>>>(cAm, cAn, cBm, cBn, Mmat);
    dim3 g3(16, 256);
    eunn_gemm<<<g3, 256, 0, stream>>>(x, Mmat, out);
}